// LossFunction_46720654246163
// MI455X (gfx1250) — compile-verified
//
#include <hip/hip_runtime.h>
#include <math.h>

typedef __attribute__((ext_vector_type(2))) float v2f;
typedef __attribute__((ext_vector_type(8))) float v8f;

#define TD 128        // feature dim
#define LDSTRIDE 132  // padded LDS row stride (dwords) to avoid bank conflicts

// ---------------------------------------------------------------------------
// Kernel 1: gather view-permuted rows, L2-normalize (norm clamped at 1e-8).
// One wave32 per row: lane loads float4 (4*32 = 128 floats).
// ---------------------------------------------------------------------------
__global__ __launch_bounds__(256) void normalize_kernel(
    const float* __restrict__ feat, float* __restrict__ fn, int N, int Bv) {
  int row  = blockIdx.x * 8 + (threadIdx.x >> 5);
  int lane = threadIdx.x & 31;
  if (row >= N) return;
  int v  = (row < Bv) ? 0 : 1;
  int bb = (row < Bv) ? row : row - Bv;
  const float4* src = (const float4*)(feat + ((size_t)bb * 2 + v) * TD) + lane;
  float4 x = *src;
  float ss = x.x * x.x + x.y * x.y + x.z * x.z + x.w * x.w;
  for (int m = 1; m < 32; m <<= 1) ss += __shfl_xor(ss, m, 32);
  float scale = 1.0f / fmaxf(sqrtf(ss), 1e-8f);
  float4 y = {x.x * scale, x.y * scale, x.z * scale, x.w * scale};
  ((float4*)(fn + (size_t)row * TD))[lane] = y;
}

// ---------------------------------------------------------------------------
// Kernel 2: fused sim-GEMM (WMMA f32 16x16x4, two independent 16-chains) +
// streaming softmax with fixed shift M = |w|+b (valid since |sim| <= 1, so
// exp(logit-M) <= 1; loss is mathematically invariant to the shift) +
// positive capture + off-diagonal argmax. All cross-lane reductions deferred
// to after the column loop: per-tile epilogue is pure per-lane VALU.
// Block = 4 waves = 64 rows; streams 16-col tiles of fn through LDS.
// ---------------------------------------------------------------------------
__global__ __launch_bounds__(128) void simloss_kernel(
    const float* __restrict__ fn, const float* __restrict__ wp,
    const float* __restrict__ bp, float* __restrict__ ploss,
    float* __restrict__ pcorr, int N, int Bv) {
  __shared__ float Bt[16 * LDSTRIDE];

  const int tid  = threadIdx.x;
  const int wave = tid >> 5;
  const int lane = tid & 31;
  const int n16  = lane & 15;   // column slot / A-row within tile
  const int h    = lane >> 4;   // K-pair selector (0: K=0,1  1: K=2,3)
  const float w  = wp[0];
  const float bs = bp[0];
  const float M  = fabsf(w) + bs;  // upper bound on all logits (|sim|<=1)

  const int rowTile = blockIdx.x * 64 + wave * 16;

  // Preload A operand: lane holds A[n16][4*kk + 2*h .. +1] for kk = 0..31
  v2f areg[32];
  const float* arow = fn + (size_t)(rowTile + n16) * TD + 2 * h;
#pragma unroll
  for (int kk = 0; kk < 32; ++kk)
    areg[kk] = *(const v2f*)(arow + 4 * kk);

  // Per-lane state; lane's C rows are m = r + 8*h.
  float rs[8], pv[8], bvv[8];
  int   bci[8], grow[8], posc[8];
#pragma unroll
  for (int r = 0; r < 8; ++r) {
    rs[r] = 0.f; pv[r] = 0.f; bvv[r] = -1e30f; bci[r] = -1;
    int g = rowTile + r + 8 * h;
    grow[r] = g;
    posc[r] = (g < Bv) ? g + Bv : g - Bv;
  }

  for (int colbase = 0; colbase < N; colbase += 16) {
    __syncthreads();
    {  // stage 16x128 column tile into LDS (128 threads x 16 floats)
      int n   = tid >> 3;   // 0..15
      int seg = tid & 7;    // 0..7
      const float4* src = (const float4*)(fn + (size_t)(colbase + n) * TD + seg * 16);
      float4* dst = (float4*)(&Bt[n * LDSTRIDE + seg * 16]);
      dst[0] = src[0]; dst[1] = src[1]; dst[2] = src[2]; dst[3] = src[3];
    }
    __syncthreads();

    // Two independent WMMA accumulation chains (K split 0..63 / 64..127)
    v8f c0 = {0.f, 0.f, 0.f, 0.f, 0.f, 0.f, 0.f, 0.f};
    v8f c1 = {0.f, 0.f, 0.f, 0.f, 0.f, 0.f, 0.f, 0.f};
    const float* brow = &Bt[n16 * LDSTRIDE + 2 * h];
#pragma unroll
    for (int kk = 0; kk < 16; ++kk) {
      v2f b0 = *(const v2f*)(brow + 4 * kk);
      v2f b1 = *(const v2f*)(brow + 4 * (kk + 16));
      c0 = __builtin_amdgcn_wmma_f32_16x16x4_f32(
          false, areg[kk], false, b0, (short)0, c0, false, false);
      c1 = __builtin_amdgcn_wmma_f32_16x16x4_f32(
          false, areg[kk + 16], false, b1, (short)0, c1, false, false);
    }
    v8f c = c0 + c1;

    const int col = colbase + n16;
#pragma unroll
    for (int r = 0; r < 8; ++r) {
      float logit  = fmaf(w, c[r], bs);
      bool  isdiag = (col == grow[r]);
      // per-lane masked exp accumulation (fixed shift M)
      float e = __expf(logit - M);
      rs[r] += isdiag ? 0.f : e;
      // positive capture: exactly one (lane, tile) matches per row
      if (col == posc[r]) pv[r] = logit;
      // per-lane running off-diagonal argmax (strict > keeps first occurrence)
      float av = isdiag ? -1e30f : logit;
      if (av > bvv[r]) { bvv[r] = av; bci[r] = col; }
    }
  }

  // One-time cross-lane reductions within 16-lane halves
  float lsum = 0.f, csum = 0.f;
#pragma unroll
  for (int r = 0; r < 8; ++r) {
    float s = rs[r];
    for (int m = 1; m < 16; m <<= 1) s += __shfl_xor(s, m, 32);
    float p = pv[r];  // all non-matching lanes hold 0
    for (int m = 1; m < 16; m <<= 1) p += __shfl_xor(p, m, 32);
    float av = bvv[r];
    int   ac = bci[r];
    for (int m = 1; m < 16; m <<= 1) {
      float ov = __shfl_xor(av, m, 32);
      int   oc = __shfl_xor(ac, m, 32);
      if (ov > av || (ov == av && oc < ac)) { av = ov; ac = oc; }
    }
    lsum += -(p - M - __logf(s));
    csum += (ac == posc[r]) ? 1.f : 0.f;
  }
  float lo = __shfl(lsum, 16, 32);  // fold lanes-16..31 half (rows 8..15)
  float co = __shfl(csum, 16, 32);
  if (lane == 0) {
    int pidx = blockIdx.x * 4 + wave;
    ploss[pidx] = lsum + lo;
    pcorr[pidx] = csum + co;
  }
}

// ---------------------------------------------------------------------------
// Kernel 3: deterministic tree reduction of partials -> (loss, prec1)
// ---------------------------------------------------------------------------
__global__ __launch_bounds__(256) void finalize_kernel(
    const float* __restrict__ pl, const float* __restrict__ pc,
    float* __restrict__ out, int P, float invN) {
  __shared__ float sl[256], sc[256];
  int t = threadIdx.x;
  float a = 0.f, b = 0.f;
  for (int i = t; i < P; i += 256) { a += pl[i]; b += pc[i]; }
  sl[t] = a; sc[t] = b;
  __syncthreads();
  for (int s = 128; s > 0; s >>= 1) {
    if (t < s) { sl[t] += sl[t + s]; sc[t] += sc[t + s]; }
    __syncthreads();
  }
  if (t == 0) {
    out[0] = sl[0] * invN;           // loss = mean over N rows
    out[1] = 100.0f * sc[0] * invN;  // prec1
  }
}

extern "C" void kernel_launch(void* const* d_in, const int* in_sizes, int n_in,
                              void* d_out, int out_size, void* d_ws, size_t ws_size,
                              hipStream_t stream) {
  const float* feat = (const float*)d_in[0];
  const float* wp   = (const float*)d_in[1];
  const float* bp   = (const float*)d_in[2];
  float* out = (float*)d_out;

  const int Dv = 128, Cv = 2;
  int Bv = in_sizes[0] / (Cv * Dv);  // 4096
  int N  = Bv * Cv;                  // 8192

  float* fn = (float*)d_ws;          // N*D floats (4 MB)
  int nBlocks2 = N / 64;             // 128
  int P = nBlocks2 * 4;              // 512 wave partials
  float* pl = fn + (size_t)N * Dv;
  float* pc = pl + P;

  normalize_kernel<<<N / 8, 256, 0, stream>>>(feat, fn, N, Bv);
  simloss_kernel<<<nBlocks2, 128, 0, stream>>>(fn, wp, bp, pl, pc, N, Bv);
  finalize_kernel<<<1, 256, 0, stream>>>(pl, pc, out, P, 1.0f / (float)N);
}